// S5Layer_55482387530211
// MI455X (gfx1250) — compile-verified
//
#include <hip/hip_runtime.h>
#include <hip/hip_bf16.h>

// ---------------------------------------------------------------------------
// S5 SSM layer for MI455X (gfx1250, wave32, WMMA).
//   D_MODEL=512, STATE=128 (HALF=64 complex modes), SEQ=4096, BATCH=8
// Pipeline: discretize -> GEMM (B_bar@u) -> parallel scan -> fused GEMM
// (2Re(C x) + u D^T) using V_WMMA_F32_16X16X4_F32 with 32x32 wave tiles.
// ---------------------------------------------------------------------------

#define DM    512
#define HALF  64
#define NSTK  128          // stacked re/im rows
#define SEQ   4096
#define BATCH 8
#define KTOT  640          // 128 (conv) + 512 (u)

typedef float v2f __attribute__((ext_vector_type(2)));
typedef float v8f __attribute__((ext_vector_type(8)));

__device__ __forceinline__ v8f wmma_f32(v2f a, v2f b, v8f c) {
  // 8 args: (neg_a, A, neg_b, B, c_mod, C, reuse_a, reuse_b)
  return __builtin_amdgcn_wmma_f32_16x16x4_f32(false, a, false, b,
                                               (short)0, c, false, false);
}

// ---- workspace layout (float offsets) -------------------------------------
#define OFF_ABAR 0u                         // 128: Abar re[64], im[64]
#define OFF_S    128u                       // 128: (Delta/denom) re[64], im[64]
#define OFF_BSTK 256u                       // 128*512
#define OFF_W    (OFF_BSTK + NSTK*DM)       // 512*640
#define OFF_V    (OFF_W + DM*KTOT)          // 8*128*4096
#define OFF_CONV (OFF_V + (size_t)BATCH*NSTK*SEQ)

// ---------------------------------------------------------------------------
// Kernel 1: bilinear discretization. 64 threads, one per complex mode.
// ---------------------------------------------------------------------------
__global__ void prep_abar(const float* __restrict__ log_Lr,
                          const float* __restrict__ Li,
                          const float* __restrict__ log_Delta,
                          float* __restrict__ ws) {
  int n = threadIdx.x;
  if (n >= HALF) return;
  float Delta = __expf(log_Delta[0]);
  float llr = fminf(fmaxf(log_Lr[n], -10.0f), 10.0f);
  float lr = -__expf(llr);
  float li = Li[n];
  // denom = 1 - (D/2) L ; num = 1 + (D/2) L
  float hr = 1.0f - 0.5f * Delta * lr, hi = -0.5f * Delta * li;
  float nr = 1.0f + 0.5f * Delta * lr, ni =  0.5f * Delta * li;
  float d  = hr * hr + hi * hi;
  float inv = 1.0f / d;
  ws[OFF_ABAR + n]        = (nr * hr + ni * hi) * inv;   // Re(Abar)
  ws[OFF_ABAR + HALF + n] = (ni * hr - nr * hi) * inv;   // Im(Abar)
  ws[OFF_S + n]           =  Delta * hr * inv;           // Re(Delta/denom)
  ws[OFF_S + HALF + n]    = -Delta * hi * inv;           // Im(Delta/denom)
}

// ---------------------------------------------------------------------------
// Kernel 2: B_bar = (Delta/denom) * (B_real + i B_imag), stacked 128x512.
// ---------------------------------------------------------------------------
__global__ void prep_bbar(const float* __restrict__ Br,
                          const float* __restrict__ Bi,
                          float* __restrict__ ws) {
  int idx = blockIdx.x * blockDim.x + threadIdx.x;
  if (idx >= HALF * DM) return;
  int n = idx / DM, p = idx - n * DM;
  float sr = ws[OFF_S + n], si = ws[OFF_S + HALF + n];
  float br = Br[idx], bi = Bi[idx];
  ws[OFF_BSTK + (size_t)n * DM + p]          = sr * br - si * bi;
  ws[OFF_BSTK + (size_t)(HALF + n) * DM + p] = sr * bi + si * br;
}

// ---------------------------------------------------------------------------
// Kernel 3: fused weight Wcat[p][k], k<64: 2*Cr, k<128: -2*Ci, else D.
// ---------------------------------------------------------------------------
__global__ void prep_w(const float* __restrict__ Cr,
                       const float* __restrict__ Ci,
                       const float* __restrict__ Dm,
                       float* __restrict__ ws) {
  int idx = blockIdx.x * blockDim.x + threadIdx.x;
  if (idx >= DM * KTOT) return;
  int p = idx / KTOT, k = idx - p * KTOT;
  float w;
  if (k < HALF)            w =  2.0f * Cr[p * HALF + k];
  else if (k < NSTK)       w = -2.0f * Ci[p * HALF + (k - HALF)];
  else                     w = Dm[p * DM + (k - NSTK)];
  ws[OFF_W + (size_t)p * KTOT + k] = w;
}

// ---------------------------------------------------------------------------
// Kernel 4: v[b][r][t] = Bstk[r][:] . u[b][t][:]   (M=128, N=T, K=512)
// One wave = one 32x32 tile (4 accumulators); 8 waves/block along t.
// ---------------------------------------------------------------------------
__global__ void __launch_bounds__(256) gemm_v(const float* __restrict__ u,
                                              const float* __restrict__ ws,
                                              float* __restrict__ wsv) {
  const int lane = threadIdx.x & 31;
  const int wave = threadIdx.x >> 5;
  const int half = lane >> 4;      // 0: K 0/1, 1: K 2/3
  const int l15  = lane & 15;
  const int t0 = (blockIdx.x * 8 + wave) * 32;
  const int m0 = blockIdx.y * 32;
  const int b  = blockIdx.z;

  const float* A0 = ws + OFF_BSTK + (size_t)(m0 + l15) * DM + half * 2;
  const float* A1 = A0 + (size_t)16 * DM;
  const float* B0 = u + ((size_t)b * SEQ + (t0 + l15)) * DM + half * 2;
  const float* B1 = B0 + (size_t)16 * DM;

  v8f acc00 = {0.f,0.f,0.f,0.f,0.f,0.f,0.f,0.f};
  v8f acc01 = acc00, acc10 = acc00, acc11 = acc00;

#pragma unroll 8
  for (int k = 0; k < DM; k += 4) {
    if ((k & 31) == 0) {
      __builtin_prefetch(B0 + k + 256, 0, 3);
      __builtin_prefetch(B1 + k + 256, 0, 3);
    }
    v2f a0 = *(const v2f*)(A0 + k);
    v2f a1 = *(const v2f*)(A1 + k);
    v2f b0 = *(const v2f*)(B0 + k);
    v2f b1 = *(const v2f*)(B1 + k);
    acc00 = wmma_f32(a0, b0, acc00);
    acc01 = wmma_f32(a0, b1, acc01);
    acc10 = wmma_f32(a1, b0, acc10);
    acc11 = wmma_f32(a1, b1, acc11);
  }

  float* out = wsv + OFF_V + ((size_t)b * NSTK) * SEQ;
#pragma unroll
  for (int j = 0; j < 8; ++j) {
    int r = m0 + j + half * 8;
    int t = t0 + l15;
    out[(size_t)r * SEQ + t]             = acc00[j];
    out[(size_t)r * SEQ + t + 16]        = acc01[j];
    out[(size_t)(r + 16) * SEQ + t]      = acc10[j];
    out[(size_t)(r + 16) * SEQ + t + 16] = acc11[j];
  }
}

// ---------------------------------------------------------------------------
// Kernel 5: parallel scan  x[t] = Abar*x[t-1] + v[t]  per (b,n).
// One block per (b,n): 256 threads x 16-step chunks, Hillis-Steele over
// chunk carries with factor Abar^(16*o), all staged through LDS.
// ---------------------------------------------------------------------------
__global__ void __launch_bounds__(256) scan_ssm(const float* __restrict__ ws,
                                                float* __restrict__ wsc) {
  __shared__ float lre[4096];
  __shared__ float lim[4096];
  __shared__ float cre[256];
  __shared__ float cim[256];

  const int tid = threadIdx.x;
  const int b = blockIdx.x >> 6;
  const int n = blockIdx.x & 63;

  const float ar = ws[OFF_ABAR + n];
  const float ai = ws[OFF_ABAR + HALF + n];

  const float* vre = ws + OFF_V + ((size_t)b * NSTK + n) * SEQ;
  const float* vim = ws + OFF_V + ((size_t)b * NSTK + HALF + n) * SEQ;
  for (int i = tid; i < SEQ; i += 256) { lre[i] = vre[i]; lim[i] = vim[i]; }
  __syncthreads();

  // pass 1: local chunk state with zero init
  const int t0 = tid * 16;
  float xr = 0.f, xi = 0.f;
#pragma unroll
  for (int i = 0; i < 16; ++i) {
    float r = ar * xr - ai * xi + lre[t0 + i];
    float m = ar * xi + ai * xr + lim[t0 + i];
    xr = r; xi = m;
  }
  cre[tid] = xr; cim[tid] = xi;
  __syncthreads();

  // Abar^16
  float Pr = ar, Pi = ai;
#pragma unroll
  for (int s = 0; s < 4; ++s) { float t = Pr * Pr - Pi * Pi; Pi = 2.f * Pr * Pi; Pr = t; }

  // inclusive scan over 256 carries
  float fr = xr, fi = xi;
  for (int o = 1; o < 256; o <<= 1) {
    float pr = 0.f, pi = 0.f;
    if (tid >= o) { pr = cre[tid - o]; pi = cim[tid - o]; }
    __syncthreads();
    fr += Pr * pr - Pi * pi;
    fi += Pr * pi + Pi * pr;
    cre[tid] = fr; cim[tid] = fi;
    __syncthreads();
    float t = Pr * Pr - Pi * Pi; Pi = 2.f * Pr * Pi; Pr = t;
  }

  float ir = 0.f, ii = 0.f;
  if (tid > 0) { ir = cre[tid - 1]; ii = cim[tid - 1]; }
  __syncthreads();

  // pass 2: recompute with carry-in, overwrite LDS with conv values
  xr = ir; xi = ii;
#pragma unroll
  for (int i = 0; i < 16; ++i) {
    float r = ar * xr - ai * xi + lre[t0 + i];
    float m = ar * xi + ai * xr + lim[t0 + i];
    xr = r; xi = m;
    lre[t0 + i] = xr; lim[t0 + i] = xi;
  }
  __syncthreads();

  float* ore = wsc + OFF_CONV + ((size_t)b * NSTK + n) * SEQ;
  float* oim = wsc + OFF_CONV + ((size_t)b * NSTK + HALF + n) * SEQ;
  for (int i = tid; i < SEQ; i += 256) { ore[i] = lre[i]; oim[i] = lim[i]; }
}

// ---------------------------------------------------------------------------
// Kernel 6: y[b][t][p] = sum_k X[bt][k] * Wcat[p][k], K = 128 (conv) + 512 (u)
// One wave = one 32(t) x 32(p) tile, 4 accumulators.
// ---------------------------------------------------------------------------
__global__ void __launch_bounds__(256) gemm_y(const float* __restrict__ u,
                                              const float* __restrict__ ws,
                                              float* __restrict__ out) {
  const int lane = threadIdx.x & 31;
  const int wave = threadIdx.x >> 5;
  const int half = lane >> 4;
  const int l15  = lane & 15;
  const int p0 = (blockIdx.x * 8 + wave) * 32;
  const int t0 = blockIdx.y * 32;
  const int b  = blockIdx.z;

  const float* conv = ws + OFF_CONV + (size_t)b * NSTK * SEQ;
  const float* W0 = ws + OFF_W + (size_t)(p0 + l15) * KTOT + half * 2;
  const float* W1 = W0 + (size_t)16 * KTOT;

  v8f acc00 = {0.f,0.f,0.f,0.f,0.f,0.f,0.f,0.f};
  v8f acc01 = acc00, acc10 = acc00, acc11 = acc00;

  // phase 1: conv region (k = 0..127), A elements strided across k
#pragma unroll 4
  for (int k = 0; k < NSTK; k += 4) {
    int kk = k + half * 2;
    const float* c0 = conv + (size_t)kk * SEQ + t0 + l15;
    v2f a0, a1;
    a0.x = c0[0];        a0.y = c0[SEQ];
    a1.x = c0[16];       a1.y = c0[SEQ + 16];
    v2f w0 = *(const v2f*)(W0 + k);
    v2f w1 = *(const v2f*)(W1 + k);
    acc00 = wmma_f32(a0, w0, acc00);
    acc01 = wmma_f32(a0, w1, acc01);
    acc10 = wmma_f32(a1, w0, acc10);
    acc11 = wmma_f32(a1, w1, acc11);
  }

  // phase 2: u region (k = 128..639), contiguous v2f loads
  const float* U0 = u + ((size_t)b * SEQ + (t0 + l15)) * DM + half * 2;
  const float* U1 = U0 + (size_t)16 * DM;
  const float* W0b = W0 + NSTK;
  const float* W1b = W1 + NSTK;
#pragma unroll 8
  for (int k = 0; k < DM; k += 4) {
    if ((k & 31) == 0) {
      __builtin_prefetch(U0 + k + 256, 0, 3);
      __builtin_prefetch(U1 + k + 256, 0, 3);
    }
    v2f a0 = *(const v2f*)(U0 + k);
    v2f a1 = *(const v2f*)(U1 + k);
    v2f w0 = *(const v2f*)(W0b + k);
    v2f w1 = *(const v2f*)(W1b + k);
    acc00 = wmma_f32(a0, w0, acc00);
    acc01 = wmma_f32(a0, w1, acc01);
    acc10 = wmma_f32(a1, w0, acc10);
    acc11 = wmma_f32(a1, w1, acc11);
  }

#pragma unroll
  for (int j = 0; j < 8; ++j) {
    int t = t0 + j + half * 8;
    int p = p0 + l15;
    out[((size_t)b * SEQ + t) * DM + p]           = acc00[j];
    out[((size_t)b * SEQ + t) * DM + p + 16]      = acc01[j];
    out[((size_t)b * SEQ + t + 16) * DM + p]      = acc10[j];
    out[((size_t)b * SEQ + t + 16) * DM + p + 16] = acc11[j];
  }
}

// ---------------------------------------------------------------------------
extern "C" void kernel_launch(void* const* d_in, const int* in_sizes, int n_in,
                              void* d_out, int out_size, void* d_ws, size_t ws_size,
                              hipStream_t stream) {
  const float* u        = (const float*)d_in[0];
  const float* log_Lr   = (const float*)d_in[1];
  const float* Li       = (const float*)d_in[2];
  const float* Br       = (const float*)d_in[3];
  const float* Bi       = (const float*)d_in[4];
  const float* Cr       = (const float*)d_in[5];
  const float* Ci       = (const float*)d_in[6];
  const float* Dm       = (const float*)d_in[7];
  const float* log_Delta= (const float*)d_in[8];
  float* ws  = (float*)d_ws;
  float* out = (float*)d_out;

  prep_abar<<<1, 64, 0, stream>>>(log_Lr, Li, log_Delta, ws);
  prep_bbar<<<(HALF * DM + 255) / 256, 256, 0, stream>>>(Br, Bi, ws);
  prep_w<<<(DM * KTOT + 255) / 256, 256, 0, stream>>>(Cr, Ci, Dm, ws);

  // v = Bstk @ u^T : 32x32 tiles, M tiles = 128/32 = 4, N tiles = 4096/32 = 128
  gemm_v<<<dim3(SEQ / 32 / 8, NSTK / 32, BATCH), 256, 0, stream>>>(u, ws, ws);

  // 512 independent scans
  scan_ssm<<<BATCH * HALF, 256, 0, stream>>>(ws, ws);

  // y : 32x32 tiles, N tiles = 512/32 = 16 (8 waves/block), M tiles = 128
  gemm_y<<<dim3(DM / 32 / 8, SEQ / 32, BATCH), 256, 0, stream>>>(u, ws, out);
}